// BridgeStructureLoss_89799176225319
// MI455X (gfx1250) — compile-verified
//
#include <hip/hip_runtime.h>
#include <math.h>

#ifndef __has_builtin
#define __has_builtin(x) 0
#endif
#if __has_builtin(__builtin_amdgcn_wmma_f32_16x16x4_f32)
#define USE_WMMA_RED 1
#else
#define USE_WMMA_RED 0
#endif

typedef __attribute__((ext_vector_type(2))) float v2f;
typedef __attribute__((ext_vector_type(8))) float v8f;

namespace {
constexpr int kB    = 32;
constexpr int kC    = 5;
constexpr int kN    = 262144;
constexpr int kBPB  = 64;              // blocks per batch
constexpr int kChunk = kN / kBPB;      // 4096 points per block
constexpr int kTPB  = 256;             // 8 waves
constexpr int kNQ   = 44;              // 36 sums + 4 mins + 4 maxs
// quantity layout per batch:
//  0..4  A[c]   = sum lse where label==c
//  5..9  B[c]   = sum logit[c] where label==c
// 10..14 T[c]   = sum logit[c] over all points
// 15..19 dist[c]= label count
// 20..23 gt count (cid 1..4)      24..27 gt sum z (cid 1..4)
// 28..31 pred count (cid 1..4)    32..35 pred sum z (cid 1..4)
// 36 mn2x 37 mn2y 38 mn3x 39 mn3y   40 mx2x 41 mx2y 42 mx3x 43 mx3y
}

__device__ inline void atomicMinF(float* addr, float val) {
  unsigned int* ai = (unsigned int*)addr;
  unsigned int cur = *ai;
  while (val < __uint_as_float(cur)) {
    unsigned int prev = atomicCAS(ai, cur, __float_as_uint(val));
    if (prev == cur) break;
    cur = prev;
  }
}

__device__ inline void atomicMaxF(float* addr, float val) {
  unsigned int* ai = (unsigned int*)addr;
  unsigned int cur = *ai;
  while (val > __uint_as_float(cur)) {
    unsigned int prev = atomicCAS(ai, cur, __float_as_uint(val));
    if (prev == cur) break;
    cur = prev;
  }
}

__device__ inline float waveSum(float v) {
#pragma unroll
  for (int o = 16; o > 0; o >>= 1) v += __shfl_xor(v, o, 32);
  return v;
}
__device__ inline float waveMin(float v) {
#pragma unroll
  for (int o = 16; o > 0; o >>= 1) v = fminf(v, __shfl_xor(v, o, 32));
  return v;
}
__device__ inline float waveMax(float v) {
#pragma unroll
  for (int o = 16; o > 0; o >>= 1) v = fmaxf(v, __shfl_xor(v, o, 32));
  return v;
}

__global__ void bsl_init(float* __restrict__ ws) {
  int i = blockIdx.x * blockDim.x + threadIdx.x;
  if (i < kB * kNQ) {
    int q = i % kNQ;
    ws[i] = (q < 36) ? 0.0f : ((q < 40) ? __builtin_inff() : -__builtin_inff());
  }
}

__global__ __launch_bounds__(kTPB)
void bsl_main(const float* __restrict__ outputs, const int* __restrict__ labels,
              const float* __restrict__ points, float* __restrict__ ws) {
  const int b     = blockIdx.x / kBPB;
  const int chunk = blockIdx.x % kBPB;
  const int n0    = chunk * kChunk;
  const float* ob = outputs + (size_t)b * kC * kN;
  const int*   lb = labels  + (size_t)b * kN;
  const float* pb = points  + (size_t)b * kN * 3;

  float acc[kNQ];
#pragma unroll
  for (int i = 0; i < 36; ++i) acc[i] = 0.0f;
#pragma unroll
  for (int i = 36; i < 40; ++i) acc[i] = __builtin_inff();
#pragma unroll
  for (int i = 40; i < 44; ++i) acc[i] = -__builtin_inff();

  for (int i = threadIdx.x; i < kChunk; i += kTPB) {
    const int n = n0 + i;
    // software prefetch of the 7 streams, 8 block-strides ahead (global_prefetch_b8)
    const int npf = n + kTPB * 8;
    if (npf < n0 + kChunk) {
#pragma unroll
      for (int c = 0; c < kC; ++c) __builtin_prefetch(ob + (size_t)c * kN + npf, 0, 1);
      __builtin_prefetch(pb + (size_t)npf * 3, 0, 1);
      __builtin_prefetch(lb + npf, 0, 1);
    }

    float l[kC];
#pragma unroll
    for (int c = 0; c < kC; ++c) l[c] = ob[(size_t)c * kN + n];
    const int   y  = lb[n];
    const float px = pb[(size_t)n * 3 + 0];
    const float py = pb[(size_t)n * 3 + 1];
    const float pz = pb[(size_t)n * 3 + 2];

    // log-sum-exp over the 5 logits
    float mx = fmaxf(fmaxf(fmaxf(l[0], l[1]), fmaxf(l[2], l[3])), l[4]);
    float es = 0.0f;
#pragma unroll
    for (int c = 0; c < kC; ++c) es += expf(l[c] - mx);
    const float lse = mx + logf(es);

    // argmax, first occurrence on ties (matches jnp.argmax)
    int pred = 0; float best = l[0];
#pragma unroll
    for (int c = 1; c < kC; ++c) { if (l[c] > best) { best = l[c]; pred = c; } }

#pragma unroll
    for (int c = 0; c < kC; ++c) {
      const bool isy = (y == c);
      acc[0 + c]  += isy ? lse  : 0.0f;
      acc[5 + c]  += isy ? l[c] : 0.0f;
      acc[10 + c] += l[c];
      acc[15 + c] += isy ? 1.0f : 0.0f;
    }
#pragma unroll
    for (int c = 1; c < kC; ++c) {
      const bool gy = (y == c), pp = (pred == c);
      acc[20 + c - 1] += gy ? 1.0f : 0.0f;
      acc[24 + c - 1] += gy ? pz   : 0.0f;
      acc[28 + c - 1] += pp ? 1.0f : 0.0f;
      acc[32 + c - 1] += pp ? pz   : 0.0f;
    }
    // masked-with-zeros bounding boxes (every point contributes 0 or its coord)
    const float x2 = (pred == 2) ? px : 0.0f, y2 = (pred == 2) ? py : 0.0f;
    const float x3 = (pred == 3) ? px : 0.0f, y3 = (pred == 3) ? py : 0.0f;
    acc[36] = fminf(acc[36], x2); acc[37] = fminf(acc[37], y2);
    acc[38] = fminf(acc[38], x3); acc[39] = fminf(acc[39], y3);
    acc[40] = fmaxf(acc[40], x2); acc[41] = fmaxf(acc[41], y2);
    acc[42] = fmaxf(acc[42], x3); acc[43] = fmaxf(acc[43], y3);
  }

  // in-wave reduction (wave32)
#pragma unroll
  for (int i = 0; i < 36; ++i) acc[i] = waveSum(acc[i]);
#pragma unroll
  for (int i = 36; i < 40; ++i) acc[i] = waveMin(acc[i]);
#pragma unroll
  for (int i = 40; i < 44; ++i) acc[i] = waveMax(acc[i]);

  __shared__ float red[8][kNQ];
  __shared__ float sfin[kNQ];
  const int wv = threadIdx.x >> 5, ln = threadIdx.x & 31;
  if (ln == 0) {
#pragma unroll
    for (int i = 0; i < kNQ; ++i) red[wv][i] = acc[i];
  }
  __syncthreads();

#if USE_WMMA_RED
  // Cross-wave sum of the 36 sum-quantities via V_WMMA_F32_16X16X4_F32.
  // B = all-ones (layout independent). A (16x4 f32, documented layout:
  // m = lane%16, k = vgpr + 2*(lane>=16)) holds quantity q's 8 wave-partials
  // in rows {2q, 2q+1}: A[2q+(w&1)][w>>1] = red[w][q]. Then the row-sum
  // D[2q][*] + D[2q+1][*] is the exact 8-wave sum. 8 quantities per WMMA.
  if (threadIdx.x < 32) {
    const int L  = threadIdx.x;
    const int hi = L >> 4;            // lane half
    const int q  = (L & 15) >> 1;     // quantity slot 0..7 this lane feeds
    const int w0 = 4 * hi + (L & 1);  // wave index for vgpr 0 (k = 2*hi)
    const int w1 = w0 + 2;            // wave index for vgpr 1 (k = 2*hi+1)
    v2f onesv; onesv[0] = 1.0f; onesv[1] = 1.0f;
#pragma unroll
    for (int g = 0; g < 5; ++g) {
      const int qq = 8 * g + q;
      v2f a;
      a[0] = (qq < 36) ? red[w0][qq] : 0.0f;
      a[1] = (qq < 36) ? red[w1][qq] : 0.0f;
      v8f cz = {0.0f, 0.0f, 0.0f, 0.0f, 0.0f, 0.0f, 0.0f, 0.0f};
      v8f d = __builtin_amdgcn_wmma_f32_16x16x4_f32(
          /*neg_a=*/false, a, /*neg_b=*/false, onesv,
          /*c_mod=*/(short)0, cz, /*reuse_a=*/false, /*reuse_b=*/false);
      // D layout: vgpr r holds M=r (lanes 0-15) / M=8+r (lanes 16-31), N=lane%16
      const int Ll = L & 15;
      if (Ll < 4) {
        const float s0 = (Ll == 0) ? (d[0] + d[1])
                       : (Ll == 1) ? (d[2] + d[3])
                       : (Ll == 2) ? (d[4] + d[5])
                       :             (d[6] + d[7]);
        const int outq = 8 * g + 4 * hi + Ll;
        if (outq < 36) sfin[outq] = s0;
      }
    }
  }
#else
  if (threadIdx.x < 36) {
    float v = 0.0f;
#pragma unroll
    for (int w = 0; w < 8; ++w) v += red[w][threadIdx.x];
    sfin[threadIdx.x] = v;
  }
#endif
  if (threadIdx.x >= 36 && threadIdx.x < 40) {
    float v = red[0][threadIdx.x];
#pragma unroll
    for (int w = 1; w < 8; ++w) v = fminf(v, red[w][threadIdx.x]);
    sfin[threadIdx.x] = v;
  } else if (threadIdx.x >= 40 && threadIdx.x < 44) {
    float v = red[0][threadIdx.x];
#pragma unroll
    for (int w = 1; w < 8; ++w) v = fmaxf(v, red[w][threadIdx.x]);
    sfin[threadIdx.x] = v;
  }
  __syncthreads();

  float* wb = ws + (size_t)b * kNQ;
  const int t = threadIdx.x;
  if (t < 36)       atomicAdd(&wb[t], sfin[t]);
  else if (t < 40)  atomicMinF(&wb[t], sfin[t]);
  else if (t < 44)  atomicMaxF(&wb[t], sfin[t]);
}

__global__ void bsl_finish(const float* __restrict__ ws, float* __restrict__ out) {
  __shared__ float wsh[kB][kC];
  const int b = threadIdx.x;
  if (b < kB) {
    const float* s = ws + (size_t)b * kNQ;
    float gtc[4], gz[4], prc[4], pzc[4];
#pragma unroll
    for (int c = 0; c < 4; ++c) {
      gtc[c] = s[20 + c];
      gz[c]  = s[24 + c] / fmaxf(gtc[c], 1.0f);
      prc[c] = s[28 + c];
      pzc[c] = s[32 + c] / fmaxf(prc[c], 1.0f);
    }
    float w[kC] = {0.8f, 2.0f, 1.5f, 1.5f, 3.5f};
    const int   pc[12] = {1, 1, 1, 2, 2, 2, 3, 3, 3, 4, 4, 4};
    const int   pr[12] = {2, 3, 4, 1, 3, 4, 1, 2, 4, 1, 2, 3};
    const float ps[12] = {-1.f, -1.f, -1.f, 1.f, -1.f, -1.f, 1.f, 1.f, -1.f, 1.f, 1.f, 1.f};
#pragma unroll
    for (int i = 0; i < 12; ++i) {
      const int cid = pc[i], ref = pr[i];
      if (prc[cid - 1] > 0.0f && gtc[ref - 1] > 0.0f) {
        const float diff = ps[i] * (pzc[cid - 1] - gz[ref - 1]);
        const float add  = 50.0f / (1.0f + expf(10.0f * diff));  // 50*sigmoid(-10*diff)
        w[cid] += add;
        w[ref] += 0.3f * add;
      }
    }
    const float mn2x = s[36], mn2y = s[37], mn3x = s[38], mn3y = s[39];
    const float mx2x = s[40], mx2y = s[41], mx3x = s[42], mx3y = s[43];
    {  // cid 2 vs box of 3
      const float dx = mx3x - mn3x + 1e-7f, dy = mx3y - mn3y + 1e-7f;
      const float xc0 = (mn2x - mn3x) / dx, xc1 = (mx2x - mn3x) / dx;
      const float yc0 = (mn2y - mn3y) / dy, yc1 = (mx2y - mn3y) / dy;
      const float v0 = (xc0 < 0.1f || xc0 > 0.9f || yc0 < 0.1f || yc0 > 0.9f) ? 1.0f : 0.0f;
      const float v1 = (xc1 < 0.1f || xc1 > 0.9f || yc1 < 0.1f || yc1 > 0.9f) ? 1.0f : 0.0f;
      w[2] += 50.0f * 0.5f * (v0 + v1);
    }
    {  // cid 3 vs box of 2
      const float dx = mx2x - mn2x + 1e-7f, dy = mx2y - mn2y + 1e-7f;
      const float xc0 = (mn3x - mn2x) / dx, xc1 = (mx3x - mn2x) / dx;
      const float yc0 = (mn3y - mn2y) / dy, yc1 = (mx3y - mn2y) / dy;
      const float v0 = (xc0 < 0.1f || xc0 > 0.9f || yc0 < 0.1f || yc0 > 0.9f) ? 1.0f : 0.0f;
      const float v1 = (xc1 < 0.1f || xc1 > 0.9f || yc1 < 0.1f || yc1 > 0.9f) ? 1.0f : 0.0f;
      w[3] += 50.0f * 0.5f * (v0 + v1);
    }
#pragma unroll
    for (int c = 0; c < kC; ++c) wsh[b][c] = fminf(fmaxf(w[c], 0.1f), 10.0f);
  }
  __syncthreads();
  if (threadIdx.x == 0) {
    double A[kC], Bv[kC], T[kC], D[kC], WM[kC];
    for (int c = 0; c < kC; ++c) { A[c] = Bv[c] = T[c] = D[c] = WM[c] = 0.0; }
    for (int bb = 0; bb < kB; ++bb) {
      const float* s = ws + (size_t)bb * kNQ;
      for (int c = 0; c < kC; ++c) {
        A[c]  += (double)s[c];
        Bv[c] += (double)s[5 + c];
        T[c]  += (double)s[10 + c];
        D[c]  += (double)s[15 + c];
        WM[c] += (double)wsh[bb][c];
      }
    }
    double cw[kC], cs = 0.0;
    for (int c = 0; c < kC; ++c) { cw[c] = 1.0 / (sqrt(D[c]) + 1e-7); cs += cw[c]; }
    double wt[kC], wsum = 0.0, nll = 0.0, sle = 0.0, sw = 0.0;
    for (int c = 0; c < kC; ++c) {
      wt[c] = (WM[c] / (double)kB) * (cw[c] / cs * (double)kC);
      wsum += wt[c] * D[c];
      nll  += wt[c] * (A[c] - Bv[c]);
      sle  += A[c];
      sw   += wt[c];
    }
    double smooth = sw * sle;
    for (int c = 0; c < kC; ++c) smooth -= wt[c] * T[c];
    const double loss = 0.8 * (nll / wsum) + 0.04 * (smooth / wsum);
    out[0] = (float)loss;
  }
}

extern "C" void kernel_launch(void* const* d_in, const int* in_sizes, int n_in,
                              void* d_out, int out_size, void* d_ws, size_t ws_size,
                              hipStream_t stream) {
  (void)in_sizes; (void)n_in; (void)out_size; (void)ws_size;
  const float* outputs = (const float*)d_in[0];
  const int*   labels  = (const int*)d_in[1];
  const float* points  = (const float*)d_in[2];
  float* out = (float*)d_out;
  float* ws  = (float*)d_ws;

  bsl_init<<<(kB * kNQ + 255) / 256, 256, 0, stream>>>(ws);
  bsl_main<<<kB * kBPB, kTPB, 0, stream>>>(outputs, labels, points, ws);
  bsl_finish<<<1, kB, 0, stream>>>(ws, out);
}